// MutualDynamics_12206297055729
// MI455X (gfx1250) — compile-verified
//
#include <hip/hip_runtime.h>

#define BLOCK 256
#define TILE_EDGES 1024
#define DIM 16

// Mutualistic dynamics constants: B=0.1, K=5, C=1, D=5, E=0.9, H=0.1
__device__ __forceinline__ float selfdyn(float v) {
    // B + x*(1 - x/K)*(x/C - 1)  with K=5 (1/K=0.2), C=1
    return 0.1f + v * (1.0f - v * 0.2f) * (v - 1.0f);
}

__global__ __launch_bounds__(BLOCK)
void self_init_kernel(const float* __restrict__ x, float* __restrict__ f, int n) {
    int i4 = (blockIdx.x * BLOCK + threadIdx.x) * 4;
    if (i4 + 4 <= n) {
        float4 v = *(const float4*)(x + i4);
        float4 r;
        r.x = selfdyn(v.x);
        r.y = selfdyn(v.y);
        r.z = selfdyn(v.z);
        r.w = selfdyn(v.w);
        *(float4*)(f + i4) = r;
    } else {
        for (int i = i4; i < n; ++i) f[i] = selfdyn(x[i]);
    }
}

__global__ __launch_bounds__(BLOCK)
void edge_kernel(const float* __restrict__ x,
                 const int* __restrict__ esrc,
                 const int* __restrict__ edst,
                 const float* __restrict__ evals,
                 float* __restrict__ f,
                 int n_edges) {
    __shared__ __align__(16) int   s_src[TILE_EDGES];
    __shared__ __align__(16) int   s_dst[TILE_EDGES];
    __shared__ __align__(16) float s_val[TILE_EDGES];

    const int t         = threadIdx.x;
    const int tile_base = blockIdx.x * TILE_EDGES;
    const int remaining = n_edges - tile_base;

    if (remaining >= TILE_EDGES) {
        // CDNA5 async Global->LDS staging of the contiguous edge metadata.
        // 256 threads x 1 b128 per array covers 1024 ints (4 KB) per array.
        unsigned l_src = (unsigned)(unsigned long long)&s_src[t * 4];
        unsigned l_dst = (unsigned)(unsigned long long)&s_dst[t * 4];
        unsigned l_val = (unsigned)(unsigned long long)&s_val[t * 4];
        const int*   g_src = esrc  + tile_base + t * 4;
        const int*   g_dst = edst  + tile_base + t * 4;
        const float* g_val = evals + tile_base + t * 4;
        asm volatile("global_load_async_to_lds_b128 %0, %1, off"
                     :: "v"(l_src), "v"(g_src) : "memory");
        asm volatile("global_load_async_to_lds_b128 %0, %1, off"
                     :: "v"(l_dst), "v"(g_dst) : "memory");
        asm volatile("global_load_async_to_lds_b128 %0, %1, off"
                     :: "v"(l_val), "v"(g_val) : "memory");
        asm volatile("s_wait_asynccnt 0" ::: "memory");
    } else {
        // Partial tail tile: plain guarded loads.
        for (int i = t; i < remaining; i += BLOCK) {
            int e = tile_base + i;
            s_src[i] = esrc[e];
            s_dst[i] = edst[e];
            s_val[i] = evals[e];
        }
    }
    __syncthreads();

    // 4 threads per edge; each handles 4 of the 16 dims as a float4.
    const int q4     = (t & 3) * 4;
    const int n_proc = remaining < TILE_EDGES ? remaining : TILE_EDGES;
    for (int el = t >> 2; el < n_proc; el += BLOCK / 4) {
        const int   src = s_src[el];
        const int   dst = s_dst[el];
        const float a   = s_val[el];
        const float4 xi = *(const float4*)(x + (size_t)src * DIM + q4);
        const float4 xj = *(const float4*)(x + (size_t)dst * DIM + q4);
        float4 m;
        m.x = a * xi.x * xj.x * __builtin_amdgcn_rcpf(5.0f + 0.9f * xi.x + 0.1f * xj.x);
        m.y = a * xi.y * xj.y * __builtin_amdgcn_rcpf(5.0f + 0.9f * xi.y + 0.1f * xj.y);
        m.z = a * xi.z * xj.z * __builtin_amdgcn_rcpf(5.0f + 0.9f * xi.z + 0.1f * xj.z);
        m.w = a * xi.w * xj.w * __builtin_amdgcn_rcpf(5.0f + 0.9f * xi.w + 0.1f * xj.w);
        float* out = f + (size_t)src * DIM + q4;
        // Native L2 fp32 atomics (segment_sum by edge_src).
        unsafeAtomicAdd(out + 0, m.x);
        unsafeAtomicAdd(out + 1, m.y);
        unsafeAtomicAdd(out + 2, m.z);
        unsafeAtomicAdd(out + 3, m.w);
    }
}

extern "C" void kernel_launch(void* const* d_in, const int* in_sizes, int n_in,
                              void* d_out, int out_size, void* d_ws, size_t ws_size,
                              hipStream_t stream) {
    (void)d_ws; (void)ws_size; (void)n_in; (void)out_size;
    const float* x     = (const float*)d_in[0];
    const int*   esrc  = (const int*)d_in[1];
    const int*   edst  = (const int*)d_in[2];
    const float* evals = (const float*)d_in[3];
    float*       f     = (float*)d_out;

    const int n_elem  = in_sizes[0];   // N_NODES * DIM
    const int n_edges = in_sizes[1];   // N_EDGES

    // Pass 1: f = self-dynamics(x)   (also initializes every output element)
    int nthr    = (n_elem + 3) / 4;
    int sblocks = (nthr + BLOCK - 1) / BLOCK;
    self_init_kernel<<<sblocks, BLOCK, 0, stream>>>(x, f, n_elem);

    // Pass 2: f[src] += A * xi * xj / (D + E*xi + H*xj)   (atomic segment_sum)
    int eblocks = (n_edges + TILE_EDGES - 1) / TILE_EDGES;
    edge_kernel<<<eblocks, BLOCK, 0, stream>>>(x, esrc, edst, evals, f, n_edges);
}